// GeneralizedNeuralODE_38001870635349
// MI455X (gfx1250) — compile-verified
//
#include <hip/hip_runtime.h>
#include <math.h>

#define B_    128
#define W_    64
#define F_    17
#define H_    256
#define RTOL_ 1e-5f
#define ATOL_ 1e-7f
#define NELEM (W_ * H_)   // 16384 per batch

typedef __attribute__((ext_vector_type(2))) float v2f;
typedef __attribute__((ext_vector_type(8))) float v8f;

// ---------------------------------------------------------------------------
// Block-wide sum; returns result to ALL threads (keeps control flow uniform).
__device__ __forceinline__ float block_reduce_sum(float v, float* sRed, int tid) {
  sRed[tid] = v;
  __syncthreads();
#pragma unroll
  for (int s = 128; s > 0; s >>= 1) {
    if (tid < s) sRed[tid] += sRed[tid + s];
    __syncthreads();
  }
  float r = sRed[0];
  __syncthreads();
  return r;
}

// ---------------------------------------------------------------------------
// vf(Z) = tanh(Z @ W1 + b1) @ W2 + b2, Z = 64x256 (LDS), result -> kout (global)
// Per-wave: N-strip of 32 cols; 4 M-tiles x 2 N-tiles of 16x16 f32 accumulators.
// K loop: 64 iterations of V_WMMA_F32_16X16X4_F32 (K=4).
__device__ __forceinline__ void vf_eval(const float* __restrict__ sZ, float* __restrict__ sU,
                                        const float* __restrict__ w1T, const float* __restrict__ w2T,
                                        const float* __restrict__ b1, const float* __restrict__ b2,
                                        float* __restrict__ kout, int tid) {
  const int lane  = tid & 31;
  const int wv    = tid >> 5;
  const int ln15  = lane & 15;
  const int hi    = lane >> 4;   // 0 or 1
  const int khalf = hi << 1;     // K offset 0 / 2 for A,B fragments
  const int moff  = hi << 3;     // M offset 0 / 8 for C,D rows
  const int ns    = wv * 32;     // this wave's N strip

  v8f acc[4][2];

  // ---- GEMM1: U = tanh(Z @ W1 + b1)
#pragma unroll
  for (int ni = 0; ni < 2; ++ni) {
    float bb = b1[ns + ni * 16 + ln15];
    v8f c;
#pragma unroll
    for (int i = 0; i < 8; ++i) c[i] = bb;
#pragma unroll
    for (int mi = 0; mi < 4; ++mi) acc[mi][ni] = c;
  }
  for (int kb = 0; kb < 64; ++kb) {
    const int kk = kb * 4 + khalf;
    v2f a[4], bf[2];
#pragma unroll
    for (int mi = 0; mi < 4; ++mi)
      a[mi] = *(const v2f*)&sZ[(mi * 16 + ln15) * 256 + kk];
#pragma unroll
    for (int ni = 0; ni < 2; ++ni)
      bf[ni] = *(const v2f*)&w1T[(ns + ni * 16 + ln15) * 256 + kk];
#pragma unroll
    for (int mi = 0; mi < 4; ++mi)
#pragma unroll
      for (int ni = 0; ni < 2; ++ni)
        acc[mi][ni] = __builtin_amdgcn_wmma_f32_16x16x4_f32(
            false, a[mi], false, bf[ni], (short)0, acc[mi][ni], false, false);
  }
#pragma unroll
  for (int mi = 0; mi < 4; ++mi)
#pragma unroll
    for (int ni = 0; ni < 2; ++ni)
#pragma unroll
      for (int i = 0; i < 8; ++i)
        sU[(mi * 16 + moff + i) * 256 + (ns + ni * 16 + ln15)] = tanhf(acc[mi][ni][i]);
  __syncthreads();

  // ---- GEMM2: kout = U @ W2 + b2
#pragma unroll
  for (int ni = 0; ni < 2; ++ni) {
    float bb = b2[ns + ni * 16 + ln15];
    v8f c;
#pragma unroll
    for (int i = 0; i < 8; ++i) c[i] = bb;
#pragma unroll
    for (int mi = 0; mi < 4; ++mi) acc[mi][ni] = c;
  }
  for (int kb = 0; kb < 64; ++kb) {
    const int kk = kb * 4 + khalf;
    v2f a[4], bf[2];
#pragma unroll
    for (int mi = 0; mi < 4; ++mi)
      a[mi] = *(const v2f*)&sU[(mi * 16 + ln15) * 256 + kk];
#pragma unroll
    for (int ni = 0; ni < 2; ++ni)
      bf[ni] = *(const v2f*)&w2T[(ns + ni * 16 + ln15) * 256 + kk];
#pragma unroll
    for (int mi = 0; mi < 4; ++mi)
#pragma unroll
      for (int ni = 0; ni < 2; ++ni)
        acc[mi][ni] = __builtin_amdgcn_wmma_f32_16x16x4_f32(
            false, a[mi], false, bf[ni], (short)0, acc[mi][ni], false, false);
  }
#pragma unroll
  for (int mi = 0; mi < 4; ++mi)
#pragma unroll
    for (int ni = 0; ni < 2; ++ni)
#pragma unroll
      for (int i = 0; i < 8; ++i)
        kout[(mi * 16 + moff + i) * 256 + (ns + ni * 16 + ln15)] = acc[mi][ni][i];
  __threadfence_block();  // make k visible to all waves in WG
  __syncthreads();
}

// ---------------------------------------------------------------------------
// sZ = y + h * (c0*k0 + ... ), thread tid owns column tid of every row.
template <int NK>
__device__ __forceinline__ void combo(float* __restrict__ sZ, const float* __restrict__ y, float h,
                                      const float* __restrict__ k0, const float* __restrict__ k1,
                                      const float* __restrict__ k2, const float* __restrict__ k3,
                                      const float* __restrict__ k4,
                                      float c0, float c1, float c2, float c3, float c4, int tid) {
  for (int j = 0; j < 64; ++j) {
    const int e = j * 256 + tid;
    float s = c0 * k0[e];
    if constexpr (NK > 1) s = fmaf(c1, k1[e], s);
    if constexpr (NK > 2) s = fmaf(c2, k2[e], s);
    if constexpr (NK > 3) s = fmaf(c3, k3[e], s);
    if constexpr (NK > 4) s = fmaf(c4, k4[e], s);
    sZ[e] = fmaf(h, s, y[e]);
  }
  __syncthreads();
}

// ---------------------------------------------------------------------------
// Kernel 1: transpose W1, W2 -> [N][K] so B-fragments are contiguous b64 loads
__global__ void __launch_bounds__(256) transpose_kernel(const float* __restrict__ w1,
                                                        const float* __restrict__ w2,
                                                        float* __restrict__ w1T,
                                                        float* __restrict__ w2T) {
  const int idx = blockIdx.x * 256 + threadIdx.x;  // 0..65535
  const float* src = blockIdx.y ? w2 : w1;
  float* dst = blockIdx.y ? w2T : w1T;
  const int k = idx >> 8, n = idx & 255;
  dst[n * 256 + k] = src[k * 256 + n];
}

// ---------------------------------------------------------------------------
// Kernel 2: per batch: t_min / t_max / rank(t[63]) and h0 = x[:,:,1:] @ w_in + b_in
__global__ void __launch_bounds__(256) setup_kernel(const float* __restrict__ x,
                                                    const float* __restrict__ w_in,
                                                    const float* __restrict__ b_in,
                                                    float* __restrict__ y,
                                                    float* __restrict__ meta) {
  const int b = blockIdx.x, tid = threadIdx.x;
  __shared__ float sx[W_ * F_];
  for (int i = tid; i < W_ * F_; i += 256) sx[i] = x[(size_t)b * W_ * F_ + i];
  __syncthreads();

  if (tid == 0) {
    const float t63 = sx[63 * F_];
    float tmn = sx[0], tmx = sx[0];
    int rank = 0;
    for (int w = 0; w < W_; ++w) {
      const float t = sx[w * F_];
      tmn = fminf(tmn, t);
      tmx = fmaxf(tmx, t);
      if (t < t63 || (t == t63 && w < 63)) rank++;  // stable argsort rank
    }
    meta[b * 4 + 0] = tmn;
    meta[b * 4 + 1] = tmx;
    meta[b * 4 + 2] = (float)rank;
  }
  const float bb = b_in[tid];
  for (int w = 0; w < W_; ++w) {
    float s = bb;
#pragma unroll
    for (int f = 0; f < F_ - 1; ++f) s = fmaf(sx[w * F_ + 1 + f], w_in[f * H_ + tid], s);
    y[(size_t)b * NELEM + w * 256 + tid] = s;
  }
}

// ---------------------------------------------------------------------------
// Kernel 3: dopri5 adaptive integration (JAX odeint semantics) + output proj.
__global__ void __launch_bounds__(256) ode_kernel(float* __restrict__ yg, float* __restrict__ kg,
                                                  const float* __restrict__ w1T,
                                                  const float* __restrict__ w2T,
                                                  const float* __restrict__ b1,
                                                  const float* __restrict__ b2,
                                                  const float* __restrict__ w_out,
                                                  const float* __restrict__ b_out,
                                                  const float* __restrict__ meta,
                                                  float* __restrict__ out) {
  __shared__ float sZ[NELEM];
  __shared__ float sU[NELEM];
  __shared__ float sRed[256];

  const int b = blockIdx.x, tid = threadIdx.x;
  float* y = yg + (size_t)b * NELEM;
  float* kb[7];
#pragma unroll
  for (int i = 0; i < 7; ++i) kb[i] = kg + ((size_t)b * 7 + i) * NELEM;
  int s1 = 0, s7 = 6;  // FSAL ping-pong slots for k1 / k7

  const float t0 = meta[b * 4 + 0];
  const float t1 = meta[b * 4 + 1];

  // dopri5 tableau
  const float A21 = 0.2f;
  const float A31 = 3.f / 40.f, A32 = 9.f / 40.f;
  const float A41 = 44.f / 45.f, A42 = -56.f / 15.f, A43 = 32.f / 9.f;
  const float A51 = 19372.f / 6561.f, A52 = -25360.f / 2187.f, A53 = 64448.f / 6561.f,
              A54 = -212.f / 729.f;
  const float A61 = 9017.f / 3168.f, A62 = -355.f / 33.f, A63 = 46732.f / 5247.f,
              A64 = 49.f / 176.f, A65 = -5103.f / 18656.f;
  const float B1 = 35.f / 384.f, B3 = 500.f / 1113.f, B4 = 125.f / 192.f,
              B5 = -2187.f / 6784.f, B6 = 11.f / 84.f;
  const float E1 = 71.f / 57600.f, E3 = -71.f / 16695.f, E4 = 71.f / 1920.f,
              E5 = -17253.f / 339200.f, E6 = 22.f / 525.f, E7 = -1.f / 40.f;

  float t = t0;
  if (t1 > t0) {
    // ---- f0 = vf(y0) -> k1
    for (int j = 0; j < 64; ++j) sZ[j * 256 + tid] = y[j * 256 + tid];
    __syncthreads();
    vf_eval(sZ, sU, w1T, w2T, b1, b2, kb[s1], tid);

    // ---- Hairer initial step size (per jax.experimental.ode)
    float l0 = 0.f, l1 = 0.f;
    for (int j = 0; j < 64; ++j) {
      const int e = j * 256 + tid;
      const float yv = y[e], f0 = kb[s1][e];
      const float sc = ATOL_ + fabsf(yv) * RTOL_;
      const float a = yv / sc, bq = f0 / sc;
      l0 += a * a;
      l1 += bq * bq;
    }
    const float d0 = sqrtf(block_reduce_sum(l0, sRed, tid));
    const float d1 = sqrtf(block_reduce_sum(l1, sRed, tid));
    const float h0i = ((d0 < 1e-5f) || (d1 < 1e-5f)) ? 1e-6f : 0.01f * d0 / d1;

    for (int j = 0; j < 64; ++j) {
      const int e = j * 256 + tid;
      sZ[e] = fmaf(h0i, kb[s1][e], y[e]);
    }
    __syncthreads();
    vf_eval(sZ, sU, w1T, w2T, b1, b2, kb[1], tid);  // f1 (temp in k2 slot)
    float l2 = 0.f;
    for (int j = 0; j < 64; ++j) {
      const int e = j * 256 + tid;
      const float sc = ATOL_ + fabsf(y[e]) * RTOL_;
      const float d = (kb[1][e] - kb[s1][e]) / sc;
      l2 += d * d;
    }
    const float d2 = sqrtf(block_reduce_sum(l2, sRed, tid)) / h0i;
    const float h1i = ((d1 <= 1e-15f) && (d2 <= 1e-15f))
                          ? fmaxf(1e-6f, h0i * 1e-3f)
                          : powf(0.01f / fmaxf(d1, d2), 1.0f / 6.0f);
    float h = fminf(100.f * h0i, h1i);

    // ---- adaptive RK45 loop (uniform control flow; WMMA needs EXEC all-1s)
    int step = 0;
    while (t < t1 && step < 4096) {
      const float hc = fminf(h, t1 - t);
      float* k1p = kb[s1];

      combo<1>(sZ, y, hc, k1p, y, y, y, y, A21, 0, 0, 0, 0, tid);
      vf_eval(sZ, sU, w1T, w2T, b1, b2, kb[1], tid);  // k2
      combo<2>(sZ, y, hc, k1p, kb[1], y, y, y, A31, A32, 0, 0, 0, tid);
      vf_eval(sZ, sU, w1T, w2T, b1, b2, kb[2], tid);  // k3
      combo<3>(sZ, y, hc, k1p, kb[1], kb[2], y, y, A41, A42, A43, 0, 0, tid);
      vf_eval(sZ, sU, w1T, w2T, b1, b2, kb[3], tid);  // k4
      combo<4>(sZ, y, hc, k1p, kb[1], kb[2], kb[3], y, A51, A52, A53, A54, 0, tid);
      vf_eval(sZ, sU, w1T, w2T, b1, b2, kb[4], tid);  // k5
      combo<5>(sZ, y, hc, k1p, kb[1], kb[2], kb[3], kb[4], A61, A62, A63, A64, A65, tid);
      vf_eval(sZ, sU, w1T, w2T, b1, b2, kb[5], tid);  // k6
      // y1 -> sZ (kept in LDS for error + accept)
      combo<5>(sZ, y, hc, k1p, kb[2], kb[3], kb[4], kb[5], B1, B3, B4, B5, B6, tid);
      vf_eval(sZ, sU, w1T, w2T, b1, b2, kb[s7], tid);  // k7 = f(y1), FSAL

      // RMS error ratio
      float le = 0.f;
      for (int j = 0; j < 64; ++j) {
        const int e = j * 256 + tid;
        float ee = E1 * k1p[e];
        ee = fmaf(E3, kb[2][e], ee);
        ee = fmaf(E4, kb[3][e], ee);
        ee = fmaf(E5, kb[4][e], ee);
        ee = fmaf(E6, kb[5][e], ee);
        ee = fmaf(E7, kb[s7][e], ee);
        ee *= hc;
        const float tol = ATOL_ + RTOL_ * fmaxf(fabsf(y[e]), fabsf(sZ[e]));
        const float rr = ee / tol;
        le += rr * rr;
      }
      const float r = sqrtf(block_reduce_sum(le, sRed, tid) * (1.0f / (float)NELEM));

      if (r <= 1.0f) {  // accept
        for (int j = 0; j < 64; ++j) {
          const int e = j * 256 + tid;
          y[e] = sZ[e];
        }
        __threadfence_block();
        __syncthreads();
        t += hc;
        const int tmp = s1;
        s1 = s7;
        s7 = tmp;
      }
      // optimal_step_size (safety 0.9, ifactor 10, dfactor 0.2, order 5)
      if (r == 0.f) {
        h = hc * 10.f;
      } else {
        const float df = (r < 1.f) ? 1.f : 0.2f;
        const float factor = fminf(10.f, fmaxf(0.9f * powf(r, -0.2f), df));
        h = hc * factor;
      }
      step++;
    }
  }

  // ---- out[b] = y[rank] . w_out + b_out
  const int rank = (int)meta[b * 4 + 2];
  const float pv = y[rank * 256 + tid] * w_out[tid];
  const float s = block_reduce_sum(pv, sRed, tid);
  if (tid == 0) out[b] = s + b_out[0];
}

// ---------------------------------------------------------------------------
extern "C" void kernel_launch(void* const* d_in, const int* in_sizes, int n_in,
                              void* d_out, int out_size, void* d_ws, size_t ws_size,
                              hipStream_t stream) {
  const float* x     = (const float*)d_in[0];
  const float* w_in  = (const float*)d_in[1];
  const float* b_in  = (const float*)d_in[2];
  const float* w1    = (const float*)d_in[3];
  const float* b1    = (const float*)d_in[4];
  const float* w2    = (const float*)d_in[5];
  const float* b2    = (const float*)d_in[6];
  const float* w_out = (const float*)d_in[7];
  const float* b_out = (const float*)d_in[8];
  float* out = (float*)d_out;

  float* ws   = (float*)d_ws;
  float* y    = ws;                                   // 128*16384
  float* kg   = y + (size_t)B_ * NELEM;               // 128*7*16384
  float* w1T  = kg + (size_t)B_ * 7 * NELEM;          // 65536
  float* w2T  = w1T + (size_t)H_ * H_;                // 65536
  float* meta = w2T + (size_t)H_ * H_;                // 128*4

  transpose_kernel<<<dim3(256, 2), 256, 0, stream>>>(w1, w2, w1T, w2T);
  setup_kernel<<<B_, 256, 0, stream>>>(x, w_in, b_in, y, meta);
  ode_kernel<<<B_, 256, 0, stream>>>(y, kg, w1T, w2T, b1, b2, w_out, b_out, meta, out);
}